// RelationalMultiAggrMP_3324304687539
// MI455X (gfx1250) — compile-verified
//
#include <hip/hip_runtime.h>
#include <hip/hip_bf16.h>

// ---- problem constants (match reference) ----
#define N_NODES   25000
#define HIDDEN    128
#define K1        256      // 2*HIDDEN
#define M3        192      // 3*MSG_DIM
#define MSG       64
#define N_TYPES   4
#define N_EDGES   100000
#define TILE_E    16
#define TILES_PER_TYPE (N_EDGES / TILE_E)   // 6250
#define THREADS   192                        // 6 waves (wave32)
#define NEG_FLT_MAX (-3.402823466e+38f)
#define TOTAL_EDGES ((long long)N_TYPES * N_EDGES)   // 400000

typedef __attribute__((ext_vector_type(2))) float v2f;
typedef __attribute__((ext_vector_type(8))) float v8f;

// D = A(16x4 f32) * B(4x16 f32) + C(16x16 f32)  -- exact fp32 WMMA on CDNA5
__device__ __forceinline__ v8f wmma_f32(v2f a, v2f b, v8f c) {
  return __builtin_amdgcn_wmma_f32_16x16x4_f32(
      /*neg_a=*/false, a, /*neg_b=*/false, b,
      /*c_mod=*/(short)0, c, /*reuse_a=*/false, /*reuse_b=*/false);
}

// one 16x16 output tile of layer-2 (h[16x192] @ W2[192x192]), K stepped by 4
__device__ __forceinline__ v8f l2_tile(const float (*sH)[196], const float* __restrict__ W2,
                                       int nt, int m16, int krow) {
  v8f acc = {};
  const int n = nt * 16 + m16;
  #pragma unroll 4
  for (int k = 0; k < M3; k += 4) {
    v2f a; a.x = sH[m16][k + krow];       a.y = sH[m16][k + krow + 1];
    v2f b; b.x = W2[(k + krow) * M3 + n]; b.y = W2[(k + krow + 1) * M3 + n];
    acc = wmma_f32(a, b, acc);
  }
  return acc;
}

__global__ void init_kernel(float* __restrict__ out, float* __restrict__ cnt) {
  long long i = (long long)blockIdx.x * blockDim.x + threadIdx.x;
  if (i < (long long)N_NODES * 256) {
    int c = (int)(i & 255);
    out[i] = (c >= 192) ? NEG_FLT_MAX : 0.0f;   // max region gets -FLT_MAX
  }
  if (i < N_NODES) cnt[i] = 0.0f;
}

// PASS 0: full messages + sum/mean-accum/max/cnt atomics (STORE: also save mean_m to ws)
// PASS 1: recompute mean third, scatter variance terms into std region (fallback path)
template <int PASS, bool STORE>
__global__ __launch_bounds__(THREADS)
void msg_kernel(const float* __restrict__ ns, const long long* __restrict__ adj,
                const float* __restrict__ W1g, const float* __restrict__ W2g,
                float* __restrict__ out, float* __restrict__ cnt,
                float* __restrict__ meanbuf) {
  __shared__ __align__(16) float sF[TILE_E][260];  // feats 16x256 (+pad, stride%64banks==4)
  __shared__ __align__(16) float sH[TILE_E][196];  // h     16x192 (+pad)
  __shared__ int sSrc[TILE_E], sTgt[TILE_E];

  const int tid = threadIdx.x;
  const int t   = blockIdx.x / TILES_PER_TYPE;
  const int e0  = (blockIdx.x % TILES_PER_TYPE) * TILE_E;

  const long long* adjT = adj + (long long)t * (2LL * N_EDGES);
  if (tid < TILE_E) {
    sSrc[tid] = (int)adjT[2 * (e0 + tid) + 0];
    sTgt[tid] = (int)adjT[2 * (e0 + tid) + 1];
  }
  __syncthreads();

  // gather feats = [ns[src] || ns[tgt]] into LDS, float4 per thread
  for (int i = tid; i < TILE_E * 64; i += THREADS) {
    const int r = i >> 6, c4 = i & 63;
    const int node = (c4 < 32) ? sSrc[r] : sTgt[r];
    const float4 v = *(const float4*)(ns + (long long)node * HIDDEN + ((c4 & 31) << 2));
    *(float4*)&sF[r][c4 << 2] = v;
  }
  __syncthreads();

  const int wave = tid >> 5;
  const int lane = tid & 31;
  const int m16  = lane & 15;
  const int hi   = lane >> 4;
  const int krow = hi * 2;   // A/B fragment: hi lanes carry K+2,K+3

  const float* W1 = W1g + (long long)t * (K1 * M3);
  const float* W2 = W2g + (long long)t * (M3 * M3);

  // ---- layer 1: h = relu(feats @ W1), 12 N-tiles, 2 per wave ----
  for (int sub = 0; sub < 2; ++sub) {
    const int nt = wave * 2 + sub;
    const int n  = nt * 16 + m16;
    v8f acc = {};
    #pragma unroll 4
    for (int k = 0; k < K1; k += 4) {
      v2f a; a.x = sF[m16][k + krow];       a.y = sF[m16][k + krow + 1];
      v2f b; b.x = W1[(k + krow) * M3 + n]; b.y = W1[(k + krow + 1) * M3 + n];
      acc = wmma_f32(a, b, acc);
    }
    #pragma unroll
    for (int r = 0; r < 8; ++r)                       // C/D: row = r + hi*8, col = m16
      sH[r + hi * 8][nt * 16 + m16] = fmaxf(acc[r], 0.0f);
  }
  __syncthreads();

  // ---- layer 2 + scatter aggregation ----
  if (PASS == 0) {
    for (int sub = 0; sub < 2; ++sub) {
      const int nt = wave * 2 + sub;
      v8f acc = l2_tile(sH, W2, nt, m16, krow);
      const int gc  = nt * 16 + m16;     // global msg column 0..191
      const int sec = gc >> 6;           // 0=sum, 1=mean, 2=max
      const int col = gc & 63;
      #pragma unroll
      for (int r = 0; r < 8; ++r) {
        const int m = r + hi * 8;
        const float v = fmaxf(acc[r], 0.0f);
        const long long base = (long long)sTgt[m] * 256;
        if (sec == 0) {
          atomicAdd(out + base + col, v);
        } else if (sec == 1) {
          if (STORE) {  // cache mean_m so the variance pass needs no GEMM recompute
            const long long g = (long long)t * N_EDGES + e0 + m;
            meanbuf[g * MSG + col] = v;
          }
          atomicAdd(out + base + 64 + col, v);                           // raw mean accum
        } else {
          atomicMax((int*)(out + base + 192 + col),                      // v>=0 vs -FLT_MAX:
                    __float_as_int(v));                                  // signed-int max exact
        }
      }
    }
    if (tid < TILE_E) atomicAdd(cnt + sTgt[tid], 1.0f);
  } else {
    // fallback variance pass: only the mean third (tiles 4..7); waves 4,5 idle here
    if (wave < 4) {
      const int nt = 4 + wave;
      v8f acc = l2_tile(sH, W2, nt, m16, krow);
      const int col = nt * 16 + m16 - 64;
      #pragma unroll
      for (int r = 0; r < 8; ++r) {
        const int m = r + hi * 8;
        const float v = fmaxf(acc[r], 0.0f);                // mean_m
        const long long base = (long long)sTgt[m] * 256;
        const float mean = out[base + 64 + col];            // finalized mean_agg
        const float var  = fmaxf(v * v - mean * mean, 0.0f) + 1e-7f;
        atomicAdd(out + base + 128 + col, var);
      }
    }
  }
}

// variance pass using cached mean_m (no GEMM): one thread per (edge, col)
__global__ void var_scatter_kernel(const long long* __restrict__ adj,
                                   const float* __restrict__ meanbuf,
                                   float* __restrict__ out) {
  const long long i = (long long)blockIdx.x * blockDim.x + threadIdx.x;
  if (i >= TOTAL_EDGES * MSG) return;
  const int c = (int)(i & 63);
  const long long g = i >> 6;                       // flat edge 0..399999
  const int t = (int)(g / N_EDGES);
  const long long e = g - (long long)t * N_EDGES;
  const int tgt = (int)adj[(long long)t * 2 * N_EDGES + 2 * e + 1];
  const float v    = meanbuf[i];
  const float mean = out[(long long)tgt * 256 + 64 + c];
  const float var  = fmaxf(v * v - mean * mean, 0.0f) + 1e-7f;
  atomicAdd(out + (long long)tgt * 256 + 128 + c, var);
}

__global__ void mean_div_kernel(float* __restrict__ out, const float* __restrict__ cnt) {
  const int i = blockIdx.x * blockDim.x + threadIdx.x;
  if (i < N_NODES * MSG) {
    const int n = i >> 6, c = i & 63;
    out[(long long)n * 256 + 64 + c] /= cnt[n];   // 0/0 -> NaN matches reference
  }
}

__global__ void sqrt_kernel(float* __restrict__ out) {
  const int i = blockIdx.x * blockDim.x + threadIdx.x;
  if (i < N_NODES * MSG) {
    const int n = i >> 6, c = i & 63;
    const long long idx = (long long)n * 256 + 128 + c;
    out[idx] = sqrtf(out[idx]);
  }
}

extern "C" void kernel_launch(void* const* d_in, const int* in_sizes, int n_in,
                              void* d_out, int out_size, void* d_ws, size_t ws_size,
                              hipStream_t stream) {
  const float*     ns  = (const float*)d_in[0];
  const long long* adj = (const long long*)d_in[1];   // int64 per reference
  const float*     W1  = (const float*)d_in[2];
  const float*     W2  = (const float*)d_in[3];
  float* out = (float*)d_out;
  float* cnt = (float*)d_ws;                          // 25000 floats of scratch

  // mean_m cache: 400000 edges x 64 f32 = 102.4 MB, placed after cnt (16B aligned)
  const size_t cnt_slots = 25024;                     // 25000 rounded up, keeps 16B align
  const size_t need = (cnt_slots + (size_t)TOTAL_EDGES * MSG) * sizeof(float);
  const bool big_ws = ws_size >= need;                // constant per process -> deterministic
  float* meanbuf = cnt + cnt_slots;

  const int init_blocks = (int)(((long long)N_NODES * 256 + 255) / 256);
  init_kernel<<<init_blocks, 256, 0, stream>>>(out, cnt);

  const int mblocks = N_TYPES * TILES_PER_TYPE;       // 25000
  const int db = (N_NODES * MSG + 255) / 256;

  if (big_ws) {
    msg_kernel<0, true><<<mblocks, THREADS, 0, stream>>>(ns, adj, W1, W2, out, cnt, meanbuf);
    mean_div_kernel<<<db, 256, 0, stream>>>(out, cnt);
    const int vb = (int)((TOTAL_EDGES * MSG + 255) / 256);
    var_scatter_kernel<<<vb, 256, 0, stream>>>(adj, meanbuf, out);
  } else {
    msg_kernel<0, false><<<mblocks, THREADS, 0, stream>>>(ns, adj, W1, W2, out, cnt, meanbuf);
    mean_div_kernel<<<db, 256, 0, stream>>>(out, cnt);
    msg_kernel<1, false><<<mblocks, THREADS, 0, stream>>>(ns, adj, W1, W2, out, cnt, meanbuf);
  }
  sqrt_kernel<<<db, 256, 0, stream>>>(out);
}